// SpikeFP32Tanh_11450382811510
// MI455X (gfx1250) — compile-verified
//
#include <hip/hip_runtime.h>
#include <stdint.h>

// SpikeFP32Tanh on MI455X (gfx1250, wave32) — TDM-staged version.
//
// 512 MB in + 512 MB out @ 23.3 TB/s -> ~46 us floor; purely bandwidth-bound.
// Input staging uses the CDNA5 Tensor Data Mover: one tensor_load_to_lds per
// 256-element (32 KB) tile, with D# padding (4 dwords per 32 dwords) so each
// element lands on a 144 B LDS stride -> 16 B-aligned ds_load_b128 rows with
// only 2-way bank conflicts. Compute: pack 32 pulses (bfe+or), FP64 tanh via
// exp(2v), unpack (bfe+cvt), non-temporal b128 stores.

typedef float        f32x4 __attribute__((ext_vector_type(4)));
typedef unsigned int u32;
typedef unsigned int u32x4 __attribute__((ext_vector_type(4)));
typedef int          i32x4 __attribute__((ext_vector_type(4)));
typedef int          i32x8 __attribute__((ext_vector_type(8)));

#define ELEMS_PER_TILE  256
#define ELEM_BYTES      128      // 32 f32 pulses per element
#define LDS_ELEM_STRIDE 144      // 128 B + 16 B TDM pad (4 dwords / 32 dwords)

__global__ __launch_bounds__(256, 1) void spike_tanh_tdm_kernel(
    const float* __restrict__ xin, float* __restrict__ xout, int n_elems)
{
    __shared__ __align__(16) char tile[ELEMS_PER_TILE * LDS_ELEM_STRIDE];

    const int base_elem = blockIdx.x * ELEMS_PER_TILE;

    // ---- wave 0 issues one TDM DMA for the whole 32 KB tile --------------
    if (threadIdx.x < 32) {
        int telems = n_elems - base_elem;
        if (telems > ELEMS_PER_TILE) telems = ELEMS_PER_TILE;
        if (telems < 0) telems = 0;
        const u32 tile_f32 = (u32)telems * 32u;   // tile_dim0 in f32 units (<= 8192)

        const uint64_t gaddr =
            (uint64_t)(uintptr_t)xin + (uint64_t)base_elem * ELEM_BYTES;
        // Flat pointers to LDS are {shared_aperture, addr32}: low 32 bits are
        // the LDS byte address the D# wants.
        const u32 lds_base = (u32)(uintptr_t)(void*)tile;

        // D# group 0: count=1 user descriptor | lds_addr | global_addr | type=2
        u32x4 g0 = {
            1u,                                     // count=1, no gather
            lds_base,                               // lds_addr (bytes)
            (u32)(gaddr & 0xFFFFFFFFu),             // global_addr[31:0]
            (u32)((gaddr >> 32) & 0x01FFFFFFu)      // global_addr[56:32]
                | 0x80000000u                       // type = 2 ("image")
        };

        // D# group 1: data_size=4B, pad 4 dwords after every 32 dwords,
        // 1-D tile: tensor_dim0 = tile_dim0 = stride = tile_f32 (no OOB).
        i32x8 g1 = {
            (int)((2u << 16)                        // data_size: 4 bytes
                | (1u << 20)                        // pad_enable
                | (4u << 22)                        // pad_interval: 32 dwords
                | (3u << 25)),                      // pad_amount: 4 dwords
            (int)(tile_f32 << 16),                  // [47:32]=0 | tensor_dim0[15:0]
            (int)((tile_f32 >> 16) | (1u << 16)),   // tensor_dim0[31:16] | tensor_dim1=1
            (int)(tile_f32 << 16),                  // tensor_dim1[31:16]=0 | tile_dim0
            0,                                      // tile_dim1=0, tile_dim2=0 (unused)
            (int)tile_f32,                          // tensor_dim0_stride[31:0]
            0,                                      // stride[47:32] | dim1_stride[15:0]
            0                                       // tensor_dim1_stride[47:16]
        };

        i32x4 gz4 = {0, 0, 0, 0};                   // groups 2/3: 1-D tile, unused
        i32x8 gz8 = {0, 0, 0, 0, 0, 0, 0, 0};       // extra group (6-arg form): zero

        __builtin_amdgcn_tensor_load_to_lds(g0, g1, gz4, gz4, gz8, 0);
        __builtin_amdgcn_s_wait_tensorcnt(0);       // TENSORcnt is per-wave
    }
    __syncthreads();                                // publish LDS to all 8 waves

    const int elem = base_elem + (int)threadIdx.x;
    if (elem < n_elems) {
        const f32x4* __restrict__ lrow = reinterpret_cast<const f32x4*>(
            &tile[(size_t)threadIdx.x * LDS_ELEM_STRIDE]);

        // ---- pack: 32 MSB-first pulses -> u32 (ds_load_b128 x8) ----------
        // Pulses are exactly 0.0f / 1.0f; bit 29 of the raw pattern is the bit.
        u32 u = 0u;
#pragma unroll
        for (int j = 0; j < 8; ++j) {
            f32x4 p = lrow[j];
            u32 b0 = (__float_as_uint(p.x) >> 29) & 1u;
            u32 b1 = (__float_as_uint(p.y) >> 29) & 1u;
            u32 b2 = (__float_as_uint(p.z) >> 29) & 1u;
            u32 b3 = (__float_as_uint(p.w) >> 29) & 1u;
            u = (u << 4) | (b0 << 3) | (b1 << 2) | (b2 << 1) | b3;
        }

        // ---- FP64 tanh exactly as the reference circuit ------------------
        float  f  = __uint_as_float(u);
        double v  = (double)f;
        double e2 = exp(2.0 * v);                   // __ocml_exp_f64, inlined
        float  t  = (float)((e2 - 1.0) / (e2 + 1.0));

        // ---- unpack: fp32 bits -> 32 MSB-first pulses, NT b128 stores ----
        u32 tu = __float_as_uint(t);
        f32x4* __restrict__ ov =
            reinterpret_cast<f32x4*>(xout) + (size_t)elem * 8;
#pragma unroll
        for (int j = 0; j < 8; ++j) {
            f32x4 q;
            q.x = (float)((tu >> (31 - (4 * j + 0))) & 1u);
            q.y = (float)((tu >> (31 - (4 * j + 1))) & 1u);
            q.z = (float)((tu >> (31 - (4 * j + 2))) & 1u);
            q.w = (float)((tu >> (31 - (4 * j + 3))) & 1u);
            __builtin_nontemporal_store(q, &ov[j]); // global_store_b128 th:NT
        }
    }
}

extern "C" void kernel_launch(void* const* d_in, const int* in_sizes, int n_in,
                              void* d_out, int out_size, void* d_ws, size_t ws_size,
                              hipStream_t stream) {
    const float* x   = (const float*)d_in[0];
    float*       out = (float*)d_out;

    // in_sizes[0] = N * 32 pulse floats
    const int n_elems = in_sizes[0] / 32;

    const int block = 256;                                   // 8 wave32
    int grid = (n_elems + ELEMS_PER_TILE - 1) / ELEMS_PER_TILE;
    if (grid < 1) grid = 1;

    spike_tanh_tdm_kernel<<<grid, block, 0, stream>>>(x, out, n_elems);
}